// GroupVQ_88210038325339
// MI455X (gfx1250) — compile-verified
//
#include <hip/hip_runtime.h>

// ---------------- problem constants ----------------
#define B_    16
#define SEQ_  8192
#define C_    192
#define H_    16
#define OV_   4
#define G_    6
#define K_    1024
#define D_    8
#define DG_   2048
#define W_    512
#define T_    128
#define ROWS_ (B_ * T_)        // 2048 (b,t) rows
#define TILES_ (ROWS_ / 16)    // 128 row-tiles per group
#define CHUNK_ 512             // K (dg) chunk
#define JCH_  (CHUNK_ / 16)    // 32 j values per chunk
#define NCH_  (DG_ / CHUNK_)   // 4 chunks

// ---------------- LDS layout (floats) ----------------
#define APITCH 260             // 16*16 + 4 pad per k-line (bank-conflict avoidance)
#define PUC_HP 289             // per-h pitch for proj_up chunk: 32*9 + 1
#define OFF_A    0             // 32 * 260            = 8320
#define OFF_BT   8320          // 512 * 8             = 4096
#define OFF_PART 12416         // 8 waves * 256       = 2048
#define OFF_PUC  8320          // union with BT+PART: 16*289 = 4624 <= 6144
#define OFF_ZE   14464         // 16*8
#define OFF_ZEN  14592         // 16*8
#define OFF_ZQ   14720         // 16*8
#define OFF_RS   14848         // 16*16
#define OFF_RI   15104         // 16*16 (ints)
#define OFF_CODE 15360         // 16 (ints)
#define SMEM_F   15376         // 61,504 bytes (< 64 KB -> always launchable)

typedef float v2f __attribute__((ext_vector_type(2)));
typedef float v8f __attribute__((ext_vector_type(8)));

// zg[b,t,g, j*16+h] lives at z_flat[seg_base(b,h,t,g) + j], j in [0,128)
__device__ __forceinline__ size_t seg_base(int b, int h, int t, int g) {
  return ((size_t)b * SEQ_ + (size_t)h * W_ + (size_t)t * OV_) * C_ + (size_t)g * 128;
}

__global__ __launch_bounds__(256) void groupvq_main(
    const float* __restrict__ z, const float* __restrict__ pd,
    const float* __restrict__ pu, const float* __restrict__ cbs,
    float* __restrict__ out, float* __restrict__ wsloss) {
  extern __shared__ float sm[];
  float* sA    = sm + OFF_A;
  float* sBt   = sm + OFF_BT;
  float* sPart = sm + OFF_PART;
  float* sPuc  = sm + OFF_PUC;
  float* sZe   = sm + OFF_ZE;
  float* sZen  = sm + OFF_ZEN;
  float* sZq   = sm + OFF_ZQ;
  float* sRs   = sm + OFF_RS;
  int*   sRi   = (int*)(sm + OFF_RI);
  int*   sCode = (int*)(sm + OFF_CODE);

  const int tid  = threadIdx.x;
  const int wave = tid >> 5;           // 8 waves (wave32)
  const int lane = tid & 31;
  const int g       = blockIdx.x / TILES_;
  const int rt      = blockIdx.x % TILES_;
  const int rowBase = rt * 16;
  const int b       = rowBase / T_;    // tile never crosses a batch boundary (16 | 128)
  const int t0      = rowBase % T_;

  const float* cbg = cbs + (size_t)g * K_ * D_;

  v8f acc = {};

  // Branchless B-fragment zero padding for columns 8..15:
  // all lanes read a valid LDS address (broadcast for the mirror lanes) and
  // multiply by a hoisted 0/1 mask -> no exec-mask branches around the WMMAs.
  const int   mm    = lane & 15;       // fragment row (A) / col (B) index
  const int   m8    = mm & 7;          // mirrored valid B column
  const float vmask = (mm < 8) ? 1.0f : 0.0f;

  // ====== down projection: z_e(16x8) = zg(16x2048) @ proj_down[g](2048x8) ======
  for (int ch = 0; ch < NCH_; ++ch) {
    const int j0  = ch * JCH_;
    const int dg0 = ch * CHUNK_;
    // stage A tile: per (m,h) one contiguous 128B piece of its 512B segment
#pragma unroll
    for (int i = 0; i < 8; ++i) {
      const int p4  = tid + i * 256;            // [0, 2048) float4 slots
      const int jj4 = p4 & 7;
      const int h   = (p4 >> 3) & 15;
      const int m   = p4 >> 7;
      const float* src = z + seg_base(b, h, t0 + m, g) + j0 + jj4 * 4;
      const float4 v = *(const float4*)src;
      if (ch + 1 < NCH_) __builtin_prefetch(src + JCH_, 0, 1); // global_prefetch_b8
      const int jl = jj4 * 4;                   // LDS: [k-line][h*16+m], K = jl*16+h
      sA[(jl + 0) * APITCH + h * 16 + m] = v.x;
      sA[(jl + 1) * APITCH + h * 16 + m] = v.y;
      sA[(jl + 2) * APITCH + h * 16 + m] = v.z;
      sA[(jl + 3) * APITCH + h * 16 + m] = v.w;
    }
    // stage B chunk: proj_down[g][dg0 : dg0+512][0:8] (d fastest, contiguous)
#pragma unroll
    for (int i = 0; i < 4; ++i) {
      const int p4 = tid + i * 256;             // [0, 1024)
      ((float4*)sBt)[p4] = ((const float4*)(pd + ((size_t)g * DG_ + dg0) * D_))[p4];
    }
    __syncthreads();

    // 8 waves split this chunk's K: wave w takes k in [w*64, w*64+64)
#pragma unroll
    for (int s = 0; s < 16; ++s) {
      const int k    = (wave << 6) + (s << 2);      // K base, multiple of 4
      const int ksel = k + ((lane >> 4) << 1);      // lanes 16-31 hold K+2,K+3
      const int jl   = ksel >> 4;
      const int hh   = ksel & 15;                   // ksel,ksel+1 share jl (hh<=14)
      v2f a;                                        // A 16x4 f32 fragment
      a.x = sA[jl * APITCH + hh * 16 + mm];
      a.y = sA[jl * APITCH + (hh + 1) * 16 + mm];
      v2f bb;                                       // B 4x16 (cols 8..15 zero-pad)
      bb.x = sBt[ksel * 8 + m8] * vmask;
      bb.y = sBt[(ksel + 1) * 8 + m8] * vmask;
      acc = __builtin_amdgcn_wmma_f32_16x16x4_f32(false, a, false, bb,
                                                  (short)0, acc, false, false);
    }
    __syncthreads();   // protect sA/sBt before next chunk's staging
  }

  // ====== cross-wave K reduction -> z_e in LDS ======
#pragma unroll
  for (int r = 0; r < 8; ++r) sPart[wave * 256 + r * 32 + lane] = acc[r];
  __syncthreads();
  if (tid < 128) {                      // (m,d): C layout VGPR r: lanes<16 M=r, >=16 M=r+8
    const int m = tid >> 3, d = tid & 7;
    const int r  = m & 7;
    const int ln = ((m >> 3) << 4) + d;
    float s = 0.f;
#pragma unroll
    for (int w = 0; w < 8; ++w) s += sPart[w * 256 + r * 32 + ln];
    sZe[m * 8 + d] = s;
  }
  __syncthreads();
  if (tid < 16) {                       // l2norm(z_e) with clip(norm, 1e-12)
    float n2 = 0.f;
#pragma unroll
    for (int d = 0; d < 8; ++d) { const float v = sZe[tid * 8 + d]; n2 += v * v; }
    const float inv = 1.0f / fmaxf(sqrtf(n2), 1e-12f);
#pragma unroll
    for (int d = 0; d < 8; ++d) sZen[tid * 8 + d] = sZe[tid * 8 + d] * inv;
  }
  __syncthreads();

  // ====== codebook argmax: 16 threads per row, 64 entries each ======
  {
    const int m = tid & 15, prt = tid >> 4;
    float zr[8];
#pragma unroll
    for (int d = 0; d < 8; ++d) zr[d] = sZen[m * 8 + d];
    float best = -3.0e38f; int bidx = prt << 6;
    const float4* cb4 = (const float4*)cbg;
    for (int kk = 0; kk < 64; ++kk) {
      const int k = (prt << 6) + kk;
      const float4 c0 = cb4[2 * k];
      const float4 c1 = cb4[2 * k + 1];
      const float dot = zr[0]*c0.x + zr[1]*c0.y + zr[2]*c0.z + zr[3]*c0.w +
                        zr[4]*c1.x + zr[5]*c1.y + zr[6]*c1.z + zr[7]*c1.w;
      const float n2  = c0.x*c0.x + c0.y*c0.y + c0.z*c0.z + c0.w*c0.w +
                        c1.x*c1.x + c1.y*c1.y + c1.z*c1.z + c1.w*c1.w;
      const float sim = dot / fmaxf(sqrtf(n2), 1e-12f);
      if (sim > best) { best = sim; bidx = k; }   // strict > keeps first max
    }
    sRs[m * 16 + prt] = best;
    sRi[m * 16 + prt] = bidx;
  }
  __syncthreads();
  if (tid < 16) {                        // ascending scan preserves argmax-first
    float bb = -3.0e38f; int bi = 0;
#pragma unroll
    for (int p = 0; p < 16; ++p) {
      const float s = sRs[tid * 16 + p];
      if (s > bb) { bb = s; bi = sRi[tid * 16 + p]; }
    }
    sCode[tid] = bi;
  }
  __syncthreads();
  if (tid < 128) sZq[tid] = cbg[(size_t)sCode[tid >> 3] * D_ + (tid & 7)];
  __syncthreads();
  // per-tile loss partial (commitment == codebook loss in forward)
  if (tid < 128) { const float df = sZe[tid] - sZq[tid]; sRs[tid] = df * df; }
  __syncthreads();
  if (tid == 0) {
    float s = 0.f;
    for (int i = 0; i < 128; ++i) s += sRs[i];
    wsloss[blockIdx.x] = s;
  }

  // ====== up projection + coalesced scatter (z_q forward == straight-through) ======
  const float* pug = pu + (size_t)g * D_ * DG_;
  for (int ch = 0; ch < NCH_; ++ch) {
    const int dg0 = ch * CHUNK_;
    const int j0  = ch * JCH_;
    __syncthreads();                     // previous-chunk readers done before overwrite
#pragma unroll
    for (int i = 0; i < 4; ++i) {
      const int p4   = tid + i * 256;    // [0, 1024) float4 of this pu chunk
      const int d    = p4 >> 7;
      const int dgl4 = (p4 & 127) * 4;
      const float4 v = *(const float4*)(pug + (size_t)d * DG_ + dg0 + dgl4);
#pragma unroll
      for (int q = 0; q < 4; ++q) {
        const int dgl = dgl4 + q;
        const int h   = dgl & 15;
        const int jl  = dgl >> 4;
        const float vv = (q == 0) ? v.x : (q == 1) ? v.y : (q == 2) ? v.z : v.w;
        sPuc[h * PUC_HP + jl * 9 + d] = vv;   // pitch 9 -> conflict-free reads
      }
    }
    __syncthreads();
    for (int seg = wave; seg < 256; seg += 8) {   // 256 (m,h) segments
      const int m = seg >> 4;
      const int h = seg & 15;
      const float q0 = sZq[m*8+0], q1 = sZq[m*8+1], q2 = sZq[m*8+2], q3 = sZq[m*8+3];
      const float q4 = sZq[m*8+4], q5 = sZq[m*8+5], q6 = sZq[m*8+6], q7 = sZq[m*8+7];
      const float* pr = sPuc + h * PUC_HP + lane * 9;
      const float val = q0*pr[0] + q1*pr[1] + q2*pr[2] + q3*pr[3] +
                        q4*pr[4] + q5*pr[5] + q6*pr[6] + q7*pr[7];
      // lanes write consecutive floats -> 128B coalesced stores
      out[seg_base(b, h, t0 + m, g) + j0 + lane] = val;
    }
  }
}

// fixed-order loss reduction -> deterministic (no float atomics)
__global__ void groupvq_loss(const float* __restrict__ ws, float* __restrict__ loss) {
  const int b = threadIdx.x;
  if (b < B_) {
    float s = 0.f;
    for (int g = 0; g < G_; ++g)
      for (int i = 0; i < 8; ++i)         // 8 row-tiles per batch element
        s += ws[g * TILES_ + b * 8 + i];
    s *= (1.0f / (float)(T_ * D_ * G_));  // mean over (T,D) then mean over G
    loss[b]      = s;                      // commitment
    loss[B_ + b] = s;                      // codebook loss (identical forward)
  }
}

extern "C" void kernel_launch(void* const* d_in, const int* in_sizes, int n_in,
                              void* d_out, int out_size, void* d_ws, size_t ws_size,
                              hipStream_t stream) {
  (void)in_sizes; (void)n_in; (void)out_size; (void)ws_size;
  const float* z  = (const float*)d_in[0];
  const float* pd = (const float*)d_in[1];
  const float* pu = (const float*)d_in[2];
  const float* cb = (const float*)d_in[3];
  float* out = (float*)d_out;
  float* ws  = (float*)d_ws;    // 768 floats of per-tile loss partials
  const size_t shmem = (size_t)SMEM_F * sizeof(float);
  groupvq_main<<<dim3(G_ * TILES_), dim3(256), shmem, stream>>>(z, pd, pu, cb, out, ws);
  groupvq_loss<<<dim3(1), dim3(32), 0, stream>>>(ws, out + (size_t)B_ * SEQ_ * C_);
}